// RelGraphConvLayer_55757265437244
// MI455X (gfx1250) — compile-verified
//
#include <hip/hip_runtime.h>
#include <math.h>

#define DIM 128

typedef __attribute__((ext_vector_type(2))) float v2f;
typedef __attribute__((ext_vector_type(8))) float v8f;

// ---------------------------------------------------------------- degrees ---
__global__ void k_deg_count(const int* __restrict__ idx, int* __restrict__ deg, int n) {
    int i = blockIdx.x * blockDim.x + threadIdx.x;
    if (i < n) atomicAdd(&deg[idx[i]], 1);
}

__global__ void k_deg_to_norm(float* __restrict__ p, int n) {
    int i = blockIdx.x * blockDim.x + threadIdx.x;
    if (i < n) {
        int d = ((const int*)p)[i];
        if (d < 1) d = 1;
        p[i] = rsqrtf((float)d);   // clip(deg,1)^-0.5
    }
}

// --------------------------------------------------------- GEMM  Y = X @ W ---
// Block: 256 threads = 8 waves; wave w owns N-tile [16w,16w+16).
// W^T staged in LDS (64 KB) so B operand is one ds_load_b64 per k-step.
// fp32 WMMA 16x16x4; 32 chained WMMAs per 16x16 output tile over K=128.
template <int STRIPS>
__global__ __launch_bounds__(256) void k_gemm_xw(const float* __restrict__ X,
                                                 const float* __restrict__ W,
                                                 float* __restrict__ Y, int M) {
    __shared__ float sWt[DIM * DIM];              // W transposed: sWt[n*128 + k]
    const int tid = threadIdx.x;
    for (int i = tid; i < DIM * DIM; i += 256) {
        int r = i >> 7, c = i & 127;
        sWt[c * DIM + r] = W[i];
    }
    __syncthreads();

    const int wave  = tid >> 5;
    const int lane  = tid & 31;
    const int lhalf = lane >> 4;                  // 0: K pair {k,k+1}, 1: {k+2,k+3}
    const int lmod  = lane & 15;                  // M index (A) / N index (B,C,D)
    const int n0    = wave * 16;

    for (int s = 0; s < STRIPS; ++s) {
        const int m0 = (blockIdx.x * STRIPS + s) * 16;
        if (m0 >= M) return;

        const float* xr = X   + (size_t)(m0 + lmod) * DIM + lhalf * 2;   // A row
        const float* wc = sWt + (size_t)(n0 + lmod) * DIM + lhalf * 2;   // B col (transposed)

        v8f acc = {0.f, 0.f, 0.f, 0.f, 0.f, 0.f, 0.f, 0.f};
#pragma unroll
        for (int k = 0; k < DIM; k += 4) {
            v2f a = *(const v2f*)(xr + k);        // global_load_b64
            v2f b = *(const v2f*)(wc + k);        // ds_load_b64
            acc = __builtin_amdgcn_wmma_f32_16x16x4_f32(
                false, a, false, b, (short)0, acc, false, false);
        }
        // C/D layout: vgpr r -> M = m0 + r + 8*lhalf, N = n0 + lmod
#pragma unroll
        for (int r = 0; r < 8; ++r)
            Y[(size_t)(m0 + r + lhalf * 8) * DIM + (n0 + lmod)] = acc[r];
    }
}

// ------------------------------------------------------------- edge scatter --
// acc[dst] += norm_src[src] * norm_dst[dst] * Y[src]   (one wave32 per edge)
__global__ __launch_bounds__(256) void k_scatter_edges(
        const float* __restrict__ Y,
        const int* __restrict__ src, const int* __restrict__ dst,
        const float* __restrict__ nsrc, const float* __restrict__ ndst,
        float* __restrict__ acc, int nedge) {
    int gw   = (blockIdx.x * blockDim.x + threadIdx.x) >> 5;
    int lane = threadIdx.x & 31;
    int nw   = (gridDim.x * blockDim.x) >> 5;
    for (int e = gw; e < nedge; e += nw) {
        int s = src[e], d = dst[e];
        float cf = nsrc[s] * ndst[d];
        const float4 v = ((const float4*)(Y + (size_t)s * DIM))[lane];  // global_load_b128
        float* ar = acc + (size_t)d * DIM + lane * 4;
        unsafeAtomicAdd(ar + 0, cf * v.x);        // global_atomic_add_f32
        unsafeAtomicAdd(ar + 1, cf * v.y);
        unsafeAtomicAdd(ar + 2, cf * v.z);
        unsafeAtomicAdd(ar + 3, cf * v.w);
    }
}

// ---------------------------------------------------- bias + relu + L2 norm --
__global__ __launch_bounds__(256) void k_finalize_rows(float* __restrict__ H,
                                                       const float* __restrict__ bias,
                                                       int nrows) {
    int gw   = (blockIdx.x * blockDim.x + threadIdx.x) >> 5;
    int lane = threadIdx.x & 31;
    int nw   = (gridDim.x * blockDim.x) >> 5;
    for (int r = gw; r < nrows; r += nw) {
        float* row = H + (size_t)r * DIM;
        float4 v = ((const float4*)row)[lane];
        float4 b = ((const float4*)bias)[lane];
        v.x = fmaxf(v.x + b.x, 0.f);
        v.y = fmaxf(v.y + b.y, 0.f);
        v.z = fmaxf(v.z + b.z, 0.f);
        v.w = fmaxf(v.w + b.w, 0.f);
        float ss = v.x * v.x + v.y * v.y + v.z * v.z + v.w * v.w;
#pragma unroll
        for (int o = 16; o > 0; o >>= 1) ss += __shfl_xor(ss, o, 32);
        float inv = 1.0f / fmaxf(sqrtf(ss), 1e-12f);
        v.x *= inv; v.y *= inv; v.z *= inv; v.w *= inv;
        ((float4*)row)[lane] = v;
    }
}

// -------------------------------------------------------------------- host ---
extern "C" void kernel_launch(void* const* d_in, const int* in_sizes, int n_in,
                              void* d_out, int out_size, void* d_ws, size_t ws_size,
                              hipStream_t stream) {
    const float* x_drug = (const float*)d_in[0];
    const float* x_gene = (const float*)d_in[1];
    const float* W_dd   = (const float*)d_in[2];
    const float* W_dg   = (const float*)d_in[3];
    const float* W_gd   = (const float*)d_in[4];
    const float* W_gg   = (const float*)d_in[5];
    const float* h_bias = (const float*)d_in[6];
    const int* src_dd = (const int*)d_in[7];
    const int* dst_dd = (const int*)d_in[8];
    const int* src_dg = (const int*)d_in[9];
    const int* dst_dg = (const int*)d_in[10];
    const int* src_gd = (const int*)d_in[11];
    const int* dst_gd = (const int*)d_in[12];
    const int* src_gg = (const int*)d_in[13];
    const int* dst_gg = (const int*)d_in[14];

    const int n_drug = in_sizes[0] / DIM;   // 100000
    const int n_gene = in_sizes[1] / DIM;   // 50000
    const int e_dd = in_sizes[7], e_dg = in_sizes[9];
    const int e_gd = in_sizes[11], e_gg = in_sizes[13];

    // workspace layout (floats)
    float* ws   = (float*)d_ws;
    float* y_dd = ws;                                     // n_drug*128
    float* y_dg = y_dd + (size_t)n_drug * DIM;            // n_drug*128
    float* y_gd = y_dg + (size_t)n_drug * DIM;            // n_gene*128
    float* y_gg = y_gd + (size_t)n_gene * DIM;            // n_gene*128
    float* nrm  = y_gg + (size_t)n_gene * DIM;            // 8 degree/norm arrays
    float* n_out_dd = nrm;
    float* n_in_dd  = n_out_dd + n_drug;
    float* n_out_dg = n_in_dd  + n_drug;
    float* n_in_dg  = n_out_dg + n_drug;
    float* n_out_gd = n_in_dg  + n_gene;
    float* n_in_gd  = n_out_gd + n_gene;
    float* n_out_gg = n_in_gd  + n_drug;
    float* n_in_gg  = n_out_gg + n_gene;
    const int n_nrm = 4 * n_drug + 4 * n_gene;

    float* out_drug = (float*)d_out;                      // [n_drug,128]
    float* out_gene = out_drug + (size_t)n_drug * DIM;    // [n_gene,128]

    // 1) zero accumulators + degree counters
    hipMemsetAsync(nrm, 0, (size_t)n_nrm * sizeof(float), stream);
    hipMemsetAsync(d_out, 0, (size_t)out_size * sizeof(float), stream);

    // 2) degree counts (int atomics into the norm slots)
    auto degN = [](int n) { return dim3((n + 255) / 256); };
    k_deg_count<<<degN(e_dd), 256, 0, stream>>>(src_dd, (int*)n_out_dd, e_dd);
    k_deg_count<<<degN(e_dd), 256, 0, stream>>>(dst_dd, (int*)n_in_dd,  e_dd);
    k_deg_count<<<degN(e_dg), 256, 0, stream>>>(src_dg, (int*)n_out_dg, e_dg);
    k_deg_count<<<degN(e_dg), 256, 0, stream>>>(dst_dg, (int*)n_in_dg,  e_dg);
    k_deg_count<<<degN(e_gd), 256, 0, stream>>>(src_gd, (int*)n_out_gd, e_gd);
    k_deg_count<<<degN(e_gd), 256, 0, stream>>>(dst_gd, (int*)n_in_gd,  e_gd);
    k_deg_count<<<degN(e_gg), 256, 0, stream>>>(src_gg, (int*)n_out_gg, e_gg);
    k_deg_count<<<degN(e_gg), 256, 0, stream>>>(dst_gg, (int*)n_in_gg,  e_gg);

    // 3) deg -> rsqrt(max(deg,1)), in place over all 8 arrays
    k_deg_to_norm<<<degN(n_nrm), 256, 0, stream>>>(nrm, n_nrm);

    // 4) unscaled GEMMs  Y = X @ W   (fp32 WMMA 16x16x4)
    constexpr int STRIPS = 16;                 // 256 rows per block
    const int gb_drug = (n_drug / 16 + STRIPS - 1) / STRIPS;
    const int gb_gene = (n_gene / 16 + STRIPS - 1) / STRIPS;
    k_gemm_xw<STRIPS><<<gb_drug, 256, 0, stream>>>(x_drug, W_dd, y_dd, n_drug);
    k_gemm_xw<STRIPS><<<gb_drug, 256, 0, stream>>>(x_drug, W_dg, y_dg, n_drug);
    k_gemm_xw<STRIPS><<<gb_gene, 256, 0, stream>>>(x_gene, W_gd, y_gd, n_gene);
    k_gemm_xw<STRIPS><<<gb_gene, 256, 0, stream>>>(x_gene, W_gg, y_gg, n_gene);

    // 5) edge scatter with fused src/dst norms, summed across etypes in d_out
    const dim3 sgrid(4096);
    k_scatter_edges<<<sgrid, 256, 0, stream>>>(y_dd, src_dd, dst_dd, n_out_dd, n_in_dd, out_drug, e_dd);
    k_scatter_edges<<<sgrid, 256, 0, stream>>>(y_gd, src_gd, dst_gd, n_out_gd, n_in_gd, out_drug, e_gd);
    k_scatter_edges<<<sgrid, 256, 0, stream>>>(y_dg, src_dg, dst_dg, n_out_dg, n_in_dg, out_gene, e_dg);
    k_scatter_edges<<<sgrid, 256, 0, stream>>>(y_gg, src_gg, dst_gg, n_out_gg, n_in_gg, out_gene, e_gg);

    // 6) bias + relu + L2 row-normalize, in place on d_out
    const int nrows = n_drug + n_gene;
    const int fblocks = (nrows * 32 + 255) / 256;
    k_finalize_rows<<<fblocks, 256, 0, stream>>>((float*)d_out, h_bias, nrows);
}